// DisneyNet_21234318312201
// MI455X (gfx1250) — compile-verified
//
#include <hip/hip_runtime.h>
#include <hip/hip_bf16.h>

#define NTHREADS 256
#define ELEMS_PER_THREAD 8
#define EPSF 1e-6f

// ---------------- LDS layout (in floats) ----------------
// Tables that need the _tri() normalization come first so one pass fixes them.
#define OFF_CGENC   0        // tabCgEnc : 2048x1   -> 2048  (tri)
#define OFF_RSENC   2048     // tabRSEnc : 64x64x3  -> 12288 (tri)
#define OFF_LVR     14336    // tabLVR   : 32x32x1  -> 1024  (tri)
#define OFF_LV      15360    // tabLV    : 32x32x4  -> 4096  (tri)
#define OFF_NHL     19456    // tabNHL   : 32x32x4  -> 4096  (tri)
#define TRI_COUNT   23552
#define OFF_RADEC   23552    // tabRADec : 48x48x3  -> 6912  (raw)
#define OFF_CGDEC   30464    // tabCgDec : 96x2     -> 192   (raw)
#define LDS_FLOATS  30656    // 122,624 bytes (fits 320KB WGP LDS, 2 blocks/WGP)

#if defined(__has_builtin)
#if __has_builtin(__builtin_amdgcn_global_load_async_to_lds_b128)
#define HAS_ASYNC_LDS 1
#endif
#endif

// 16-byte int vector type matching the async-to-LDS builtin's parameter type.
typedef int v4i_t __attribute__((vector_size(16)));
typedef __attribute__((address_space(1))) v4i_t gv4i_t;  // global (__device__)
typedef __attribute__((address_space(3))) v4i_t lv4i_t;  // LDS (__shared__)

__device__ __forceinline__ float clamp01(float x) {
    return fminf(fmaxf(x, 0.0f), 1.0f - EPSF);
}
// tri(x) = 2*|x/2 - floor(x/2 + 0.5)|
__device__ __forceinline__ float trif(float x) {
    return 2.0f * fabsf(x * 0.5f - floorf(x * 0.5f + 0.5f));
}

// Stage nfloats (multiple of 4, 16B-aligned both sides) from global to LDS.
__device__ __forceinline__ void stage16(const float* __restrict__ g, float* l,
                                        int nfloats, int tid) {
    const int n4 = nfloats >> 2;
    for (int i = tid; i < n4; i += NTHREADS) {
#ifdef HAS_ASYNC_LDS
        // CDNA5 async bulk copy: GLOBAL_LOAD_ASYNC_TO_LDS_B128 (ASYNCcnt path)
        __builtin_amdgcn_global_load_async_to_lds_b128(
            (gv4i_t*)(g + (i << 2)),
            (lv4i_t*)(l + (i << 2)),
            /*offset=*/0, /*cpol=*/0);
#else
        float4 v = reinterpret_cast<const float4*>(g)[i];
        reinterpret_cast<float4*>(l)[i] = v;
#endif
    }
}

__device__ __forceinline__ void wait_async_lds() {
#ifdef HAS_ASYNC_LDS
#if __has_builtin(__builtin_amdgcn_s_wait_asynccnt)
    __builtin_amdgcn_s_wait_asynccnt(0);
#else
    asm volatile("s_wait_asynccnt 0" ::: "memory");
#endif
#endif
}

// Bilinear 2D lookup on a UxVxC LDS-resident table; uv already in [0,1].
template <int U, int V, int C>
__device__ __forceinline__ void bilin2d(const float* __restrict__ t,
                                        float uu, float vv, float* out) {
    float u = clamp01(uu) * (float)(U - 1);
    float v = clamp01(vv) * (float)(V - 1);
    float u0f = floorf(u), v0f = floorf(v);
    float fu = u - u0f, fv = v - v0f;
    int u0 = (int)u0f, v0 = (int)v0f;
    int u1 = min(u0 + 1, U - 1);
    int v1 = min(v0 + 1, V - 1);
    const float* p00 = t + (u0 * V + v0) * C;
    const float* p10 = t + (u1 * V + v0) * C;
    const float* p01 = t + (u0 * V + v1) * C;
    const float* p11 = t + (u1 * V + v1) * C;
    float w00 = (1.0f - fu) * (1.0f - fv);
    float w10 = fu * (1.0f - fv);
    float w01 = (1.0f - fu) * fv;
    float w11 = fu * fv;
#pragma unroll
    for (int c = 0; c < C; ++c)
        out[c] = p00[c] * w00 + p10[c] * w10 + p01[c] * w01 + p11[c] * w11;
}

template <int U, int C>
__device__ __forceinline__ void bilin1d(const float* __restrict__ t,
                                        float uu, float* out) {
    float u = clamp01(uu) * (float)(U - 1);
    float u0f = floorf(u);
    float fu = u - u0f;
    int u0 = (int)u0f;
    int u1 = min(u0 + 1, U - 1);
#pragma unroll
    for (int c = 0; c < C; ++c)
        out[c] = t[u0 * C + c] * (1.0f - fu) + t[u1 * C + c] * fu;
}

__global__ void __launch_bounds__(NTHREADS, 2)
disney_lut_kernel(const float2* __restrict__ roughNoh,
                  const float2* __restrict__ anisoToh,
                  const float2* __restrict__ nDotLV,
                  const float2* __restrict__ metalLoh,
                  const float2* __restrict__ subCg,
                  const float2* __restrict__ spst,
                  const float2* __restrict__ shst,
                  const float2* __restrict__ lumCs,
                  const float* __restrict__ tabRAEnc,   // 2048x2048x2 (global/L2)
                  const float* __restrict__ tabCgEnc,
                  const float* __restrict__ tabRSEnc,
                  const float* __restrict__ tabRADec,
                  const float* __restrict__ tabCgDec,
                  const float* __restrict__ tabLVR,
                  const float* __restrict__ tabLV,
                  const float* __restrict__ tabNHL,
                  float2* __restrict__ out, int n) {
    __shared__ float lds[LDS_FLOATS];
    const int tid = threadIdx.x;

    // ---- Stage all small tables into LDS via async-to-LDS DMA ----
    stage16(tabCgEnc, lds + OFF_CGENC, 2048, tid);
    stage16(tabRSEnc, lds + OFF_RSENC, 12288, tid);
    stage16(tabLVR,   lds + OFF_LVR,   1024, tid);
    stage16(tabLV,    lds + OFF_LV,    4096, tid);
    stage16(tabNHL,   lds + OFF_NHL,   4096, tid);
    stage16(tabRADec, lds + OFF_RADEC, 6912, tid);
    stage16(tabCgDec, lds + OFF_CGDEC, 192,  tid);
    wait_async_lds();
    __syncthreads();

    // Apply _tri() once per texel instead of per lookup.
    for (int i = tid; i < TRI_COUNT; i += NTHREADS)
        lds[i] = trif(lds[i]);
    __syncthreads();

    const long long base = (long long)blockIdx.x * (NTHREADS * ELEMS_PER_THREAD);
#pragma unroll
    for (int k = 0; k < ELEMS_PER_THREAD; ++k) {
        long long e = base + (long long)k * NTHREADS + tid;
        if (e >= n) continue;

        const float2 rN  = roughNoh[e];
        const float2 aT  = anisoToh[e];
        const float2 nlv = nDotLV[e];
        const float2 mL  = metalLoh[e];
        const float2 sCg = subCg[e];
        const float2 sp  = spst[e];
        const float2 sh  = shst[e];
        const float2 lc  = lumCs[e];

        // ra = tri(tabRAEnc)[round(u), round(v)]  (nearest, 32MB table in L2)
        int ui = (int)rintf(clamp01(rN.x) * 2047.0f);
        int vi = (int)rintf(clamp01(aT.x) * 2047.0f);
        const float2 raRaw =
            reinterpret_cast<const float2*>(tabRAEnc)[ui * 2048 + vi];
        float ra0 = trif(raRaw.x);
        float ra1 = trif(raRaw.y);

        // cg = tri(tabCgEnc)[round(.)]  (nearest; tri pre-applied in LDS)
        int ci = (int)rintf(clamp01(sCg.y) * 2047.0f);
        float cg = lds[OFF_CGENC + ci];

        // rsEnc = bilinear on tri(tabRSEnc) at (roughNoh.x, subCg.x)
        float rs[3];
        bilin2d<64, 64, 3>(lds + OFF_RSENC, rN.x, sCg.x, rs);

        float met  = mL.x;
        float omm  = 1.0f - met;
        float cd   = omm * lc.x;
        float cm0  = sp.x * 0.1f * omm * sp.y + met * lc.x;
        float cm1  = sp.x * 0.1f * omm * (1.0f - sp.y);
        float cs0  = sh.x * omm * sh.y;
        float cs1  = sh.x * omm * (1.0f - sh.y);
        float cc   = 0.0625f * lc.y;  // 0.25 * 0.25
        float rsCd0 = rs[0] * cd, rsCd1 = rs[1] * cd, rsCd2 = rs[2] * cd;

        float rc[3];
        bilin2d<48, 48, 3>(lds + OFF_RADEC, ra0, ra1, rc);
        float cgc[2];
        bilin1d<96, 2>(lds + OFF_CGDEC, cg, cgc);
        float vt;
        bilin2d<32, 32, 1>(lds + OFF_LVR, nlv.x * nlv.y, ra0, &vt);
        float nlvc[4];
        bilin2d<32, 32, 4>(lds + OFF_LV, nlv.x, nlv.y, nlvc);
        float nhlc[4];
        bilin2d<32, 32, 4>(lds + OFF_NHL, mL.y, nlv.x, nhlc);

        float noh2  = rN.y * rN.y;
        float den   = rc[0] * aT.y * aT.y + rc[1] * noh2 + rc[2];
        float dTerm = 0.25f / (nlv.y * den * den);

        float x = rsCd0 * mL.y * nlvc[1] + rsCd1 * nlvc[2] + rsCd2 * nlvc[3]
                + cm0 * vt * nhlc[2] * dTerm + cs0 * nhlc[0];
        float y = ((1.0f - cm1) * nhlc[3] + cm1) * vt * dTerm
                + cs1 * nhlc[0]
                + cc * nhlc[1] * nlvc[0] / (nlv.y * (cgc[0] * noh2 + cgc[1]));

        out[e] = make_float2(x, y);
    }
}

extern "C" void kernel_launch(void* const* d_in, const int* in_sizes, int n_in,
                              void* d_out, int out_size, void* d_ws, size_t ws_size,
                              hipStream_t stream) {
    (void)n_in; (void)out_size; (void)d_ws; (void)ws_size;
    const int n = in_sizes[0] / 2;  // (N,2) streams

    const float2* roughNoh = (const float2*)d_in[0];
    const float2* anisoToh = (const float2*)d_in[1];
    const float2* nDotLV   = (const float2*)d_in[2];
    // d_in[3] = tDotLV : unused by the reference
    const float2* metalLoh = (const float2*)d_in[4];
    const float2* subCg    = (const float2*)d_in[5];
    const float2* spst     = (const float2*)d_in[6];
    const float2* shst     = (const float2*)d_in[7];
    const float2* lumCs    = (const float2*)d_in[8];
    const float* tabRAEnc  = (const float*)d_in[9];
    const float* tabCgEnc  = (const float*)d_in[10];
    const float* tabRSEnc  = (const float*)d_in[11];
    const float* tabRADec  = (const float*)d_in[12];
    const float* tabCgDec  = (const float*)d_in[13];
    const float* tabLVR    = (const float*)d_in[14];
    const float* tabLV     = (const float*)d_in[15];
    const float* tabNHL    = (const float*)d_in[16];

    const int elemsPerBlock = NTHREADS * ELEMS_PER_THREAD;
    const int blocks = (n + elemsPerBlock - 1) / elemsPerBlock;

    disney_lut_kernel<<<blocks, NTHREADS, 0, stream>>>(
        roughNoh, anisoToh, nDotLV, metalLoh, subCg, spst, shst, lumCs,
        tabRAEnc, tabCgEnc, tabRSEnc, tabRADec, tabCgDec, tabLVR, tabLV, tabNHL,
        (float2*)d_out, n);
}